// Gaussians_41480794145202
// MI455X (gfx1250) — compile-verified
//
#include <hip/hip_runtime.h>

#define G_EPS 1e-8f
#define TILE  256
#define ITER  8
#define NWAVE (TILE / 32)

#if __has_builtin(__builtin_amdgcn_global_load_async_to_lds_b128) && \
    __has_builtin(__builtin_amdgcn_s_wait_asynccnt)
#define HAVE_ASYNC 1
#else
#define HAVE_ASYNC 0
#endif

// Async-copy builtin operand types: pointer to 16-byte int vector, in the
// global (addrspace 1) / LDS (addrspace 3) address spaces respectively.
typedef int v4i __attribute__((vector_size(4 * sizeof(int))));
typedef __attribute__((address_space(1))) v4i GV4;  // global int4*pointee
typedef __attribute__((address_space(3))) v4i LV4;  // LDS    int4 pointee

// Per-element math: R from (unnormalized) quaternion, S = |s|+eps,
// O = (R*diag(S)) * (R*diag(S))^T.  Normalization folded into inv2 = 2/|q|^2.
__device__ __forceinline__ void rssr(float w, float x, float y, float z,
                                     float s0, float s1, float s2, float* o) {
  float n2   = w * w + x * x + y * y + z * z;
  float inv2 = 2.0f / n2;
  float tx = inv2 * x, ty = inv2 * y, tz = inv2 * z, tw = inv2 * w;
  float xx = tx * x, yy = ty * y, zz = tz * z;
  float xy = tx * y, xz = tx * z, yz = ty * z;
  float wx = tw * x, wy = tw * y, wz = tw * z;

  float R00 = 1.0f - (yy + zz), R01 = xy - wz,          R02 = xz + wy;
  float R10 = xy + wz,          R11 = 1.0f - (xx + zz), R12 = yz - wx;
  float R20 = xz - wy,          R21 = yz + wx,          R22 = 1.0f - (xx + yy);

  s0 = fabsf(s0) + G_EPS;
  s1 = fabsf(s1) + G_EPS;
  s2 = fabsf(s2) + G_EPS;

  float a0 = R00 * s0, a1 = R01 * s1, a2 = R02 * s2;
  float b0 = R10 * s0, b1 = R11 * s1, b2 = R12 * s2;
  float c0 = R20 * s0, c1 = R21 * s1, c2 = R22 * s2;

  o[0] = a0 * a0 + a1 * a1 + a2 * a2;
  o[1] = a0 * b0 + a1 * b1 + a2 * b2;
  o[2] = a0 * c0 + a1 * c1 + a2 * c2;
  o[3] = o[1];
  o[4] = b0 * b0 + b1 * b1 + b2 * b2;
  o[5] = b0 * c0 + b1 * c1 + b2 * c2;
  o[6] = o[2];
  o[7] = o[5];
  o[8] = c0 * c0 + c1 * c1 + c2 * c2;
}

__global__ __launch_bounds__(TILE) void gaussians_rssr_kernel(
    const float* __restrict__ quat, const float* __restrict__ scale,
    float* __restrict__ out, long long N) {
  const int tid  = threadIdx.x;
  const int lane = tid & 31;
  const int wave = tid >> 5;
  const long long base = (long long)blockIdx.x * (TILE * (long long)ITER);

  // Per-wave output transpose tile: 32 elems * 9 floats. gcd(9,64)=1 ->
  // both write (lane*9+k) and read (lane+32k) patterns are bank-conflict-free.
  __shared__ float sO[NWAVE][32 * 9];

  if (base + (long long)(TILE * ITER) <= N) {
#if HAVE_ASYNC
    __shared__ __align__(16) float4 sQ[2][TILE];        // 8 KB
    __shared__ __align__(16) float  sS[2][TILE * 3];    // 6 KB

    // Wave-local async staging: 2 async instructions per wave per tile.
    auto prefetch = [&](int t, int buf) {
      const float* gq = quat + (base + (long long)t * TILE + tid) * 4;
      __builtin_amdgcn_global_load_async_to_lds_b128(
          (GV4*)gq, (LV4*)&sQ[buf][tid], 0, 0);
      if (lane < 24) {  // 24 lanes * 16B = this wave's 32 elements * 12B
        const float* gs =
            scale + (base + (long long)t * TILE + wave * 32) * 3 + lane * 4;
        __builtin_amdgcn_global_load_async_to_lds_b128(
            (GV4*)gs, (LV4*)&sS[buf][wave * 96 + lane * 4], 0, 0);
      }
    };

    prefetch(0, 0);
    for (int t = 0; t < ITER; ++t) {
      const int buf = t & 1;
      if (t + 1 < ITER) {
        prefetch(t + 1, buf ^ 1);
        // 4 outstanding (2 per tile); async loads retire oldest-first,
        // so waiting to <=2 guarantees tile t is resident in LDS.
        __builtin_amdgcn_s_wait_asynccnt(2);
      } else {
        __builtin_amdgcn_s_wait_asynccnt(0);
      }

      float4 q  = sQ[buf][tid];
      float  s0 = sS[buf][tid * 3 + 0];
      float  s1 = sS[buf][tid * 3 + 1];
      float  s2 = sS[buf][tid * 3 + 2];
      float  o[9];
      rssr(q.x, q.y, q.z, q.w, s0, s1, s2, o);

#pragma unroll
      for (int k = 0; k < 9; ++k) sO[wave][lane * 9 + k] = o[k];
      // Wave-local LDS ops are in-order (DScnt) -> no barrier needed.
      float* gout = out + (base + (long long)t * TILE + wave * 32) * 9;
#pragma unroll
      for (int k = 0; k < 9; ++k)
        __builtin_nontemporal_store(sO[wave][lane + 32 * k],
                                    gout + lane + 32 * k);
    }
#else
    // Sync fallback: still coalesced b128 quat loads + LDS store transpose.
    for (int t = 0; t < ITER; ++t) {
      const long long i = base + (long long)t * TILE + tid;
      float4 q  = ((const float4*)quat)[i];
      float  s0 = scale[i * 3 + 0];
      float  s1 = scale[i * 3 + 1];
      float  s2 = scale[i * 3 + 2];
      float  o[9];
      rssr(q.x, q.y, q.z, q.w, s0, s1, s2, o);

#pragma unroll
      for (int k = 0; k < 9; ++k) sO[wave][lane * 9 + k] = o[k];
      float* gout = out + (base + (long long)t * TILE + wave * 32) * 9;
#pragma unroll
      for (int k = 0; k < 9; ++k)
        __builtin_nontemporal_store(sO[wave][lane + 32 * k],
                                    gout + lane + 32 * k);
    }
#endif
  } else {
    // Tail block: simple guarded per-element path.
    for (int t = 0; t < ITER; ++t) {
      const long long i = base + (long long)t * TILE + tid;
      if (i < N) {
        float4 q  = ((const float4*)quat)[i];
        float  s0 = scale[i * 3 + 0];
        float  s1 = scale[i * 3 + 1];
        float  s2 = scale[i * 3 + 2];
        float  o[9];
        rssr(q.x, q.y, q.z, q.w, s0, s1, s2, o);
#pragma unroll
        for (int k = 0; k < 9; ++k) out[i * 9 + k] = o[k];
      }
    }
  }
}

extern "C" void kernel_launch(void* const* d_in, const int* in_sizes, int n_in,
                              void* d_out, int out_size, void* d_ws,
                              size_t ws_size, hipStream_t stream) {
  const float* quat  = (const float*)d_in[0];
  const float* scale = (const float*)d_in[1];
  float*       out   = (float*)d_out;

  const long long N         = (long long)in_sizes[0] / 4;  // (N,4) quats
  const long long per_block = (long long)TILE * ITER;
  const int       blocks    = (int)((N + per_block - 1) / per_block);

  gaussians_rssr_kernel<<<blocks, TILE, 0, stream>>>(quat, scale, out, N);
}